// CALayer_43447889166408
// MI455X (gfx1250) — compile-verified
//
#include <hip/hip_runtime.h>
#include <math.h>

typedef __attribute__((ext_vector_type(2))) float v2f;
typedef __attribute__((ext_vector_type(8))) float v8f;

#define NEG_SLOPE 0.2f

// ---------------------------------------------------------------------------
// Kernel 1: global average pool. One block per (b,c) plane (2048 planes).
// Each plane = 16384 f32 = 4096 float4. 256 threads * 16 float4 each.
// ---------------------------------------------------------------------------
__global__ __launch_bounds__(256) void ca_pool_kernel(const float* __restrict__ x,
                                                      float* __restrict__ pooled) {
    const int plane = blockIdx.x;                       // b*256 + c
    const float4* p4 = (const float4*)(x + (size_t)plane * 16384);
    const int tid = threadIdx.x;

    float s = 0.f;
#pragma unroll
    for (int i = 0; i < 16; ++i) {
        float4 v = p4[tid + i * 256];
        s += (v.x + v.y) + (v.z + v.w);
    }
    // wave32 reduction
#pragma unroll
    for (int off = 16; off > 0; off >>= 1)
        s += __shfl_xor(s, off, 32);

    __shared__ float ws[8];
    const int wave = tid >> 5, lane = tid & 31;
    if (lane == 0) ws[wave] = s;
    __syncthreads();
    if (tid == 0) {
        float t = 0.f;
#pragma unroll
        for (int w = 0; w < 8; ++w) t += ws[w];
        pooled[plane] = t * (1.0f / 16384.0f);
    }
}

// ---------------------------------------------------------------------------
// Kernel 2: squeeze-excite MLP on one wave using V_WMMA_F32_16X16X4_F32.
//   Stage y (8x256, zero-padded to 16x256) in LDS so every fragment load is
//   an unconditional ds_load_b64 / global_load_b64 — no EXEC divergence in
//   the WMMA loops.
//   GEMM1: D1[16x16] = Ypad[16x256] x W1^T[256x16]   (64 K-steps)
//   bias + LeakyReLU -> LDS (re-layout D -> A fragment)
//   GEMM2: per 16-col tile of C: D2[16x16] = Y1[16x16] x W2tile^T[16x16]
//   bias + sigmoid -> gates[8][256]
// Fragment layouts (ISA 7.12.2, 32-bit, wave32):
//   A: lane L -> M = L&15, K = k0 + 2*(L>>4) + {0,1} in the two VGPRs
//   B: lane L -> N = L&15, same K split
//   D: VGPR j, lane n<16 -> (M=j, N=n); lane n>=16 -> (M=j+8, N=n-16)
// ---------------------------------------------------------------------------
__global__ __launch_bounds__(32) void ca_se_kernel(const float* __restrict__ pooled,
                                                   const float* __restrict__ w1,
                                                   const float* __restrict__ b1,
                                                   const float* __restrict__ w2,
                                                   const float* __restrict__ b2,
                                                   float* __restrict__ gates) {
    const int lane = threadIdx.x;        // 0..31, single full wave -> EXEC all 1s
    const int m    = lane & 15;          // M (GEMM A row) or N (GEMM B col)
    const int hi   = lane >> 4;          // half-wave selects K pair

    __shared__ __align__(16) float ys[16 * 256];   // y zero-padded to 16 rows
    __shared__ __align__(16) float y1s[16 * 16];   // y1 as [m][s]

    // ---- stage pooled into LDS, zero rows 8..15 ----
    {
        float4* ys4 = (float4*)ys;
        const float4* p4 = (const float4*)pooled;
#pragma unroll
        for (int i = 0; i < 16; ++i)               // 512 float4 of real data
            ys4[lane + i * 32] = p4[lane + i * 32];
        const float4 z = {0.f, 0.f, 0.f, 0.f};
#pragma unroll
        for (int i = 0; i < 16; ++i)               // 512 float4 of zero pad
            ys4[512 + lane + i * 32] = z;
    }
    __syncthreads();

    // ---- GEMM1: Ypad[16x256] x W1^T ----
    v8f acc = {};
    for (int k0 = 0; k0 < 256; k0 += 4) {
        const int kb = k0 + hi * 2;
        const v2f a = *(const v2f*)&ys[m * 256 + kb];       // ds_load_b64
        const v2f b = *(const v2f*)(w1 + m * 256 + kb);     // global_load_b64
        acc = __builtin_amdgcn_wmma_f32_16x16x4_f32(
            false, a, false, b, (short)0, acc, false, false);
    }

    // ---- bias + LeakyReLU, spill D1 to LDS as y1[m][s] ----
    {
        const float bias = b1[m];        // N = m for this lane
#pragma unroll
        for (int j = 0; j < 8; ++j) {
            const int row = j + hi * 8;  // D row held by this half-wave
            float v = acc[j] + bias;
            v = (v >= 0.0f) ? v : NEG_SLOPE * v;
            y1s[row * 16 + m] = v;
        }
    }
    __syncthreads();

    // ---- reload y1 in A-fragment layout (K = s, 16 total -> 4 steps) ----
    v2f a2[4];
#pragma unroll
    for (int i = 0; i < 4; ++i) {
        const int kb = i * 4 + hi * 2;
        a2[i] = *(const v2f*)&y1s[m * 16 + kb];
    }

    // ---- GEMM2 over 16 tiles of the C=256 output dimension ----
    for (int t = 0; t < 16; ++t) {
        const int c = t * 16 + m;        // N = m -> output channel
        v8f acc2 = {};
#pragma unroll
        for (int i = 0; i < 4; ++i) {
            const int kb = i * 4 + hi * 2;
            const v2f b = *(const v2f*)(w2 + c * 16 + kb);  // global_load_b64
            acc2 = __builtin_amdgcn_wmma_f32_16x16x4_f32(
                false, a2[i], false, b, (short)0, acc2, false, false);
        }
        if (hi == 0) {                   // rows 0..7 = real batches, lanes 0..15
            const float bias = b2[c];
#pragma unroll
            for (int j = 0; j < 8; ++j) {
                const float v = acc2[j] + bias;
                gates[j * 256 + c] = 1.0f / (1.0f + __expf(-v));
            }
        }
    }
}

// ---------------------------------------------------------------------------
// Kernel 3: out = x * gate[plane]. Pure bandwidth: b128 grid-stride.
// 8,388,608 float4 total; 4096 float4 per plane.
// ---------------------------------------------------------------------------
__global__ __launch_bounds__(256) void ca_scale_kernel(const float* __restrict__ x,
                                                       const float* __restrict__ gates,
                                                       float* __restrict__ out) {
    const size_t n4 = (size_t)8 * 256 * 16384 / 4;     // 8,388,608
    const size_t stride = (size_t)gridDim.x * blockDim.x;
    const float4* x4 = (const float4*)x;
    float4* o4 = (float4*)out;

    for (size_t i = (size_t)blockIdx.x * blockDim.x + threadIdx.x; i < n4; i += stride) {
        const size_t plane = i >> 12;                  // (b*256 + c)
        const float g = gates[plane];
        float4 v = x4[i];
        v.x *= g; v.y *= g; v.z *= g; v.w *= g;
        o4[i] = v;
    }
}

extern "C" void kernel_launch(void* const* d_in, const int* in_sizes, int n_in,
                              void* d_out, int out_size, void* d_ws, size_t ws_size,
                              hipStream_t stream) {
    const float* x  = (const float*)d_in[0];
    const float* w1 = (const float*)d_in[1];
    const float* b1 = (const float*)d_in[2];
    const float* w2 = (const float*)d_in[3];
    const float* b2 = (const float*)d_in[4];
    float* out = (float*)d_out;

    float* pooled = (float*)d_ws;        // 2048 f32
    float* gates  = pooled + 2048;       // 2048 f32

    ca_pool_kernel<<<2048, 256, 0, stream>>>(x, pooled);
    ca_se_kernel<<<1, 32, 0, stream>>>(pooled, w1, b1, w2, b2, gates);
    ca_scale_kernel<<<4096, 256, 0, stream>>>(x, gates, out);
}